// GeneralMechanismODE_66726611910975
// MI455X (gfx1250) — compile-verified
//
#include <hip/hip_runtime.h>

typedef __attribute__((ext_vector_type(2))) float v2f;
typedef __attribute__((ext_vector_type(8))) float v8f;

// Ordered bi-bi mechanism RHS.
// Species: [E, EA, EQ, EAB, A, B, P, Q] = y[0..7]
// v0 = kf0*E*A   - kr0*EA
// v1 = kf1*EA*B  - kr1*EAB
// v2 = kf2*EAB   - kr2*EQ*P
// v3 = kf3*EQ    - kr3*E*Q
// dydt = v_net . S  computed as  D = S^T (16x4, rows 8..15 = 0)  x  v_net^T (4x16)
// via V_WMMA_F32_16X16X4_F32: lane L<16 ends up holding dydt[row L, species 0..7]
// contiguously in D[0..7] -> two coalesced b128 stores per row.
__global__ __launch_bounds__(256) void ode_rhs_wmma(
    const float* __restrict__ y,
    const float* __restrict__ kf,
    const float* __restrict__ kr,
    float* __restrict__ out,
    int n)
{
    const int  lane = threadIdx.x & 31;
    const bool hi   = lane >= 16;       // high half supplies K=2,3 of A and B
    const int  s    = lane & 15;        // A-matrix row (species) / B-matrix col (batch row)

    const int wave  = blockIdx.x * (blockDim.x >> 5) + (threadIdx.x >> 5);
    const int nWave = gridDim.x * (blockDim.x >> 5);
    const int nTile = (n + 15) >> 4;

    // A = S^T, rows 8..15 zero.  Branchless from packed +1/-1 position masks:
    // S[0] = [-1, 1, 0, 0,-1, 0, 0, 0] -> pos 0x0002 neg 0x0011   (K=0, lanes<16, v0)
    // S[1] = [ 0,-1, 0, 1, 0,-1, 0, 0] -> pos 0x0008 neg 0x0022   (K=1, lanes<16, v1)
    // S[2] = [ 0, 0, 1,-1, 0, 0, 1, 0] -> pos 0x0044 neg 0x0008   (K=2, lanes>=16, v0)
    // S[3] = [ 1, 0,-1, 0, 0, 0, 0, 1] -> pos 0x0081 neg 0x0004   (K=3, lanes>=16, v1)
    const unsigned pos0 = hi ? 0x0044u : 0x0002u;
    const unsigned neg0 = hi ? 0x0008u : 0x0011u;
    const unsigned pos1 = hi ? 0x0081u : 0x0008u;
    const unsigned neg1 = hi ? 0x0004u : 0x0022u;
    v2f A;
    A.x = (float)((int)((pos0 >> s) & 1u) - (int)((neg0 >> s) & 1u));
    A.y = (float)((int)((pos1 >> s) & 1u) - (int)((neg1 >> s) & 1u));

    for (int tile = wave; tile < nTile; tile += nWave) {
        const int m  = tile * 16 + s;                  // batch row this lane feeds
        const int mc = (m < n) ? m : (n - 1);          // clamp: keeps EXEC all-1s for WMMA

        // y row: 32 bytes, two coalesced b128 loads (both lane halves hit same lines).
        const float4 ya = *(const float4*)(y + (size_t)mc * 8);      // E, EA, EQ, EAB
        const float4 yb = *(const float4*)(y + (size_t)mc * 8 + 4);  // A, B, P, Q

        // Each half loads only its two rate pairs (b64).
        const int ko = hi ? 2 : 0;
        const float2 kfp = *(const float2*)(kf + (size_t)mc * 4 + ko);
        const float2 krp = *(const float2*)(kr + (size_t)mc * 4 + ko);

        // B-matrix (v_net^T): lane<16 -> k=0 (v0), k=1 (v1); lane>=16 -> k=2, k=3.
        v2f Bm;
        Bm.x = hi ? (kfp.x * ya.w - krp.x * ya.z * yb.z)    // v2 = kf2*EAB - kr2*EQ*P
                  : (kfp.x * ya.x * yb.x - krp.x * ya.y);   // v0 = kf0*E*A - kr0*EA
        Bm.y = hi ? (kfp.y * ya.z - krp.y * ya.x * yb.w)    // v3 = kf3*EQ  - kr3*E*Q
                  : (kfp.y * ya.y * yb.y - krp.y * ya.w);   // v1 = kf1*EA*B - kr1*EAB

        v8f C = {0.f, 0.f, 0.f, 0.f, 0.f, 0.f, 0.f, 0.f};
        // (neg_a, A, neg_b, B, c_mod, C, reuse_a, reuse_b)
        v8f D = __builtin_amdgcn_wmma_f32_16x16x4_f32(
            false, A, false, Bm, (short)0, C, false, false);

        // Lane L<16, VGPR j = dydt[row tile*16+L, species j], j=0..7.
        if (!hi && m < n) {
            float4 o0, o1;
            o0.x = D[0]; o0.y = D[1]; o0.z = D[2]; o0.w = D[3];
            o1.x = D[4]; o1.y = D[5]; o1.z = D[6]; o1.w = D[7];
            float* p = out + (size_t)m * 8;
            *(float4*)(p)     = o0;
            *(float4*)(p + 4) = o1;
        }
    }
}

extern "C" void kernel_launch(void* const* d_in, const int* in_sizes, int n_in,
                              void* d_out, int out_size, void* d_ws, size_t ws_size,
                              hipStream_t stream) {
    // setup_inputs order: t (1), y (B*8), forward_rates (B*4), reverse_rates (B*4)
    const float* y  = (const float*)d_in[1];
    const float* kf = (const float*)d_in[2];
    const float* kr = (const float*)d_in[3];
    float* out = (float*)d_out;

    const int n = in_sizes[1] / 8;          // batch size B
    const int nTile = (n + 15) >> 4;        // 16 rows per wave-tile
    const int wavesPerBlock = 256 / 32;     // wave32: 8 waves per 256-thread block

    int blocks = (nTile + wavesPerBlock - 1) / wavesPerBlock;
    if (blocks < 1) blocks = 1;
    if (blocks > 32768) blocks = 32768;     // grid-stride loop covers the rest

    ode_rhs_wmma<<<blocks, 256, 0, stream>>>(y, kf, kr, out, n);
}